// GCNBlock_4561255268773
// MI455X (gfx1250) — compile-verified
//
#include <hip/hip_runtime.h>
#include <cstddef>

typedef __attribute__((ext_vector_type(2))) float v2f;
typedef __attribute__((ext_vector_type(8))) float v8f;

// Hardware no-return FP32 atomic add, GVS form: saddr (uniform base, SGPR pair)
// + 32-bit VGPR byte offset. Acks at L2 (Y is L2-resident on 192MB L2).
__device__ __forceinline__ void atomic_fadd_hw(float* base, int byteOff, float v) {
  asm volatile("global_atomic_add_f32 %0, %1, %2"
               :
               : "v"(byteOff), "v"(v), "s"(base)
               : "memory");
}

// ---------------------------------------------------------------------------
// Graph-structure precompute (layer invariant)
// ---------------------------------------------------------------------------
__global__ __launch_bounds__(256) void k_zero(float* __restrict__ p, int count) {
  int i = blockIdx.x * 256 + threadIdx.x;
  if (i < count) p[i] = 0.0f;
}

__global__ __launch_bounds__(256) void k_deg(const long long* __restrict__ dst,
                                             float* __restrict__ deg, int E) {
  int i = blockIdx.x * 256 + threadIdx.x;
  if (i < E) atomic_fadd_hw(deg, ((int)dst[i]) << 2, 1.0f);
}

// deg -> dinv = rsqrt(deg + 1)   (+1 = self loop; always > 0)
__global__ __launch_bounds__(256) void k_rsqrt(float* __restrict__ p, int count) {
  int i = blockIdx.x * 256 + threadIdx.x;
  if (i < count) p[i] = rsqrtf(p[i] + 1.0f);
}

__global__ __launch_bounds__(256) void k_norm(const long long* __restrict__ src,
                                              const long long* __restrict__ dst,
                                              const float* __restrict__ dinv,
                                              float* __restrict__ norm, int E) {
  int i = blockIdx.x * 256 + threadIdx.x;
  if (i < E) norm[i] = dinv[(int)src[i]] * dinv[(int)dst[i]];
}

// ---------------------------------------------------------------------------
// Fused dual GEMM: XWc = act(X)@Wc, XWt = act(X)@Wt,
// Y = bc + bt + dinvC^2*XWc + dinvT^2*XWt   (bias + self-loop message)
// One wave = one 16x16 tile of both outputs, V_WMMA_F32_16X16X4_F32.
// Block = 256 threads = 8 waves = all 8 column tiles of one 16-row stripe.
// RELU is a template parameter: no per-element select in the inner loop.
// ---------------------------------------------------------------------------
template <int RELU>
__global__ __launch_bounds__(256) void k_gemm_dual(
    const float* __restrict__ X, const float* __restrict__ Wc,
    const float* __restrict__ Wt, const float* __restrict__ bc,
    const float* __restrict__ bt, const float* __restrict__ dinvC,
    const float* __restrict__ dinvT, float* __restrict__ XWc,
    float* __restrict__ XWt, float* __restrict__ Y, int n) {
  const int lane  = threadIdx.x & 31;
  const int wave  = threadIdx.x >> 5;   // 0..7 -> N tile
  const int mBase = blockIdx.x * 16;
  const int nBase = wave * 16;
  const int half  = lane >> 4;          // 0: lanes 0-15, 1: lanes 16-31
  const int l15   = lane & 15;
  const int col   = nBase + l15;

  int arow = mBase + l15;               // clamp keeps EXEC all-ones for WMMA
  if (arow >= n) arow = n - 1;
  const float* __restrict__ xrow = X + (size_t)arow * 128;

  v8f accC = {};
  v8f accT = {};

#pragma unroll 4
  for (int k = 0; k < 128; k += 4) {
    const int kk = k + half * 2;        // lanes 16-31 carry K+2, K+3
#if __has_builtin(__builtin_amdgcn_wmma_f32_16x16x4_f32)
    float2 af = *(const float2*)(xrow + kk);
    if (RELU) { af.x = fmaxf(af.x, 0.f); af.y = fmaxf(af.y, 0.f); }
    v2f a;  a.x = af.x;  a.y = af.y;

    v2f b0; b0.x = Wc[kk * 128 + col]; b0.y = Wc[(kk + 1) * 128 + col];
    v2f b1; b1.x = Wt[kk * 128 + col]; b1.y = Wt[(kk + 1) * 128 + col];

    accC = __builtin_amdgcn_wmma_f32_16x16x4_f32(false, a, false, b0,
                                                 (short)0, accC, false, false);
    accT = __builtin_amdgcn_wmma_f32_16x16x4_f32(false, a, false, b1,
                                                 (short)0, accT, false, false);
#else
    // Scalar fallback with identical output mapping.
#pragma unroll
    for (int r = 0; r < 8; ++r) {
      int row = mBase + r + 8 * half; if (row >= n) row = n - 1;
      const float* xr = X + (size_t)row * 128;
      float s0 = 0.f, s1 = 0.f;
#pragma unroll
      for (int q = 0; q < 4; ++q) {
        float xv = xr[k + q];
        if (RELU) xv = fmaxf(xv, 0.f);
        s0 += xv * Wc[(k + q) * 128 + col];
        s1 += xv * Wt[(k + q) * 128 + col];
      }
      accC[r] += s0; accT[r] += s1;
    }
    (void)kk;
#endif
  }

  const float bsum = bc[col] + bt[col];
#pragma unroll
  for (int r = 0; r < 8; ++r) {
    const int row = mBase + r + 8 * half;   // C/D layout: M = r + 8*half
    if (row < n) {
      const size_t idx = (size_t)row * 128 + col;
      const float cv = accC[r], tv = accT[r];
      XWc[idx] = cv;
      XWt[idx] = tv;
      const float dc = dinvC[row], dt = dinvT[row];
      Y[idx] = bsum + dc * dc * cv + dt * dt * tv;
    }
  }
}

// ---------------------------------------------------------------------------
// Edge scatter: Y[dst] += norm[e] * XW[src].  One wave per edge, 4 floats/lane.
// Hardware global_atomic_add_f32 (no-return) -> RMW resolves in L2.
// ---------------------------------------------------------------------------
__global__ __launch_bounds__(256) void k_scatter(
    const long long* __restrict__ src, const long long* __restrict__ dst,
    const float* __restrict__ norm, const float* __restrict__ XW,
    float* __restrict__ Y, int E) {
  const int e = blockIdx.x * 8 + (threadIdx.x >> 5);
  if (e >= E) return;
  const int lane = threadIdx.x & 31;
  const int s = (int)src[e];
  const int d = (int)dst[e];
  const float w = norm[e];
  const float4 v = *(const float4*)(XW + (size_t)s * 128 + lane * 4);
  const int boff = d * 512 + lane * 16;   // byte offset into Y (< 2^31)
  atomic_fadd_hw(Y, boff +  0, v.x * w);
  atomic_fadd_hw(Y, boff +  4, v.y * w);
  atomic_fadd_hw(Y, boff +  8, v.z * w);
  atomic_fadd_hw(Y, boff + 12, v.w * w);
}

// ---------------------------------------------------------------------------
extern "C" void kernel_launch(void* const* d_in, const int* in_sizes, int n_in,
                              void* d_out, int out_size, void* d_ws,
                              size_t ws_size, hipStream_t stream) {
  const float*     x  = (const float*)d_in[0];
  const long long* ei = (const long long*)d_in[1];
  const long long* ci = (const long long*)d_in[2];
  const float*     Wc = (const float*)d_in[3];
  const float*     bc = (const float*)d_in[4];
  const float*     Wt = (const float*)d_in[5];
  const float*     bt = (const float*)d_in[6];

  const int d  = 128;
  const int n  = in_sizes[0] / d;
  const int E  = in_sizes[1] / 2;
  const int Ec = in_sizes[2] / 2;

  const long long* srcC = ei;       const long long* dstC = ei + E;
  const long long* srcT = ci;       const long long* dstT = ci + Ec;

  float* ws = (float*)d_ws;
  size_t o = 0;
  float* xwC   = ws + o; o += (size_t)n * d;
  float* xwT   = ws + o; o += (size_t)n * d;
  float* actA  = ws + o; o += (size_t)n * d;
  float* actB  = ws + o; o += (size_t)n * d;
  float* dinvC = ws + o; o += (size_t)n;     // dinvT adjacent -> zero as one span
  float* dinvT = ws + o; o += (size_t)n;
  float* normC = ws + o; o += (size_t)E;
  float* normT = ws + o; o += (size_t)Ec;
  if (ws_size < o * sizeof(float)) return;   // workspace too small: bail

  auto cdiv = [](int a, int b) { return (a + b - 1) / b; };

  // --- structure precompute (once; layer invariant) ---
  k_zero <<<cdiv(2 * n, 256), 256, 0, stream>>>(dinvC, 2 * n);
  k_deg  <<<cdiv(E, 256),     256, 0, stream>>>(dstC, dinvC, E);
  k_deg  <<<cdiv(Ec, 256),    256, 0, stream>>>(dstT, dinvT, Ec);
  k_rsqrt<<<cdiv(2 * n, 256), 256, 0, stream>>>(dinvC, 2 * n);
  k_norm <<<cdiv(E, 256),     256, 0, stream>>>(srcC, dstC, dinvC, normC, E);
  k_norm <<<cdiv(Ec, 256),    256, 0, stream>>>(srcT, dstT, dinvT, normT, Ec);

  // --- 4 layers; ReLU fused into next layer's A load; last layer -> d_out ---
  const float* cur = x;
  float* outBuf[4] = { actA, actB, actA, (float*)d_out };
  for (int i = 0; i < 4; ++i) {
    float* y = outBuf[i];
    if (i == 0) {
      k_gemm_dual<0><<<cdiv(n, 16), 256, 0, stream>>>(
          cur, Wc + (size_t)i * d * d, Wt + (size_t)i * d * d,
          bc + (size_t)i * d, bt + (size_t)i * d,
          dinvC, dinvT, xwC, xwT, y, n);
    } else {
      k_gemm_dual<1><<<cdiv(n, 16), 256, 0, stream>>>(
          cur, Wc + (size_t)i * d * d, Wt + (size_t)i * d * d,
          bc + (size_t)i * d, bt + (size_t)i * d,
          dinvC, dinvT, xwC, xwT, y, n);
    }
    k_scatter<<<cdiv(E, 8),  256, 0, stream>>>(srcC, dstC, normC, xwC, y, E);
    k_scatter<<<cdiv(Ec, 8), 256, 0, stream>>>(srcT, dstT, normT, xwT, y, Ec);
    cur = y;
  }
}